// DTWLoss_29626684408092
// MI455X (gfx1250) — compile-verified
//
#include <hip/hip_runtime.h>
#include <hip/hip_bf16.h>

typedef float v2f __attribute__((ext_vector_type(2)));
typedef float v8f __attribute__((ext_vector_type(8)));

#define NROW 4096
#define NCOL 4096
#define DIMK 64
#define BIGF 1e30f

// ---------------------------------------------------------------------------
// Phase 0: row norms  xx[i] = |x_i|^2, yy[j] = |y_j|^2
// ---------------------------------------------------------------------------
__global__ __launch_bounds__(256)
void dtw_norms_kernel(const float* __restrict__ x, const float* __restrict__ y,
                      float* __restrict__ xx, float* __restrict__ yy) {
    int t = blockIdx.x * blockDim.x + threadIdx.x;
    if (t >= NROW + NCOL) return;
    const float* p = (t < NROW) ? (x + (size_t)t * DIMK)
                                : (y + (size_t)(t - NROW) * DIMK);
    float s = 0.0f;
#pragma unroll
    for (int k = 0; k < DIMK; ++k) s = fmaf(p[k], p[k], s);
    if (t < NROW) xx[t] = s; else yy[t - NROW] = s;
}

// ---------------------------------------------------------------------------
// Phase 1: cost matrix via V_WMMA_F32_16X16X4_F32.
// One wave32 per 16x16 tile of c; K=64 done as 16 chained K=4 WMMAs.
//   A (16x4 f32): lane L holds M = L&15, K = 2*(L>>4) + {0,1}  (2 VGPRs)
//   B (4x16 f32): lane L holds N = L&15, K = 2*(L>>4) + {0,1}  (B = y^T)
//   C/D (16x16 f32, 8 VGPRs): VGPR v -> M = v + 8*(L>=16), N = L&15
// ---------------------------------------------------------------------------
__global__ __launch_bounds__(256)
void dtw_cost_wmma_kernel(const float* __restrict__ x, const float* __restrict__ y,
                          const float* __restrict__ xx, const float* __restrict__ yy,
                          float* __restrict__ c) {
    const int lane = threadIdx.x & 31;
    const int wave = threadIdx.x >> 5;
    const int tile = blockIdx.x * 8 + wave;          // 256*256 tiles total
    const int tm = tile >> 8;                        // tile row
    const int tn = tile & 255;                       // tile col
    const int r  = lane & 15;
    const int kh = lane >> 4;                        // 0: K={0,1}, 1: K={2,3}

    const float* xrow = x + (size_t)(tm * 16 + r) * DIMK + 2 * kh;
    const float* yrow = y + (size_t)(tn * 16 + r) * DIMK + 2 * kh;

    v8f acc = {};
#pragma unroll
    for (int kb = 0; kb < DIMK; kb += 4) {
        v2f a = *(const v2f*)(xrow + kb);
        v2f b = *(const v2f*)(yrow + kb);
        acc = __builtin_amdgcn_wmma_f32_16x16x4_f32(
            /*neg_a=*/false, a, /*neg_b=*/false, b,
            /*c_mod=*/(short)0, acc, /*reuse_a=*/false, /*reuse_b=*/false);
    }

    const int   nn    = tn * 16 + r;
    const float yn    = yy[nn];
    const int   mbase = tm * 16 + (kh << 3);         // +8 for lanes 16..31
#pragma unroll
    for (int v = 0; v < 8; ++v) {
        int   mm = mbase + v;
        float sq = xx[mm] + yn - 2.0f * acc[v];
        c[(size_t)mm * NCOL + nn] = sqrtf(fmaxf(sq, 1e-12f));
    }
}

// ---------------------------------------------------------------------------
// Min-plus scan body shared by both DTW kernels.
// Each element is an affine tropical map f(v) = min(m, v + s);
// combine(e1 earlier, e2 later): m = min(m2, m1 + s2), s = s1 + s2.
// ---------------------------------------------------------------------------
__device__ __forceinline__
float dtw_row_scan(float cx, float cy, float cz, float cw,
                   float* Dp, float* wm, float* wsum,
                   int t, int lane, int wv, int j0, bool first_row) {
    float b0, b1, b2, b3;
    if (first_row) {
        b0 = (j0 == 0) ? cx : BIGF;  b1 = BIGF;  b2 = BIGF;  b3 = BIGF;
    } else {
        float dm1 = (j0 == 0) ? BIGF : Dp[j0 - 1];
        float d0 = Dp[j0], d1 = Dp[j0 + 1], d2 = Dp[j0 + 2], d3 = Dp[j0 + 3];
        b0 = fminf(dm1 + 2.0f * cx, d0 + cx);
        b1 = fminf(d0  + 2.0f * cy, d1 + cy);
        b2 = fminf(d1  + 2.0f * cz, d2 + cz);
        b3 = fminf(d2  + 2.0f * cw, d3 + cw);
    }
    // serial inclusive min-plus over the thread's 4 elements
    float m0 = b0,                  s0 = cx;
    float m1 = fminf(b1, m0 + cy),  s1 = s0 + cy;
    float m2 = fminf(b2, m1 + cz),  s2 = s1 + cz;
    float m3 = fminf(b3, m2 + cw),  s3 = s2 + cw;

    // wave32 inclusive scan of thread aggregates
    float m = m3, s = s3;
#pragma unroll
    for (int d = 1; d < 32; d <<= 1) {
        float pm = __shfl_up(m, d);
        float ps = __shfl_up(s, d);
        if (lane >= d) { m = fminf(m, pm + s); s = ps + s; }
    }
    if (lane == 31) { wm[wv] = m; wsum[wv] = s; }
    __syncthreads();
    if (t < 32) {                                    // wave 0 scans 32 wave totals
        float qm = wm[t], qs = wsum[t];
#pragma unroll
        for (int d = 1; d < 32; d <<= 1) {
            float pm = __shfl_up(qm, d);
            float ps = __shfl_up(qs, d);
            if (t >= d) { qm = fminf(qm, pm + qs); qs = ps + qs; }
        }
        wm[t] = qm; wsum[t] = qs;
    }
    __syncthreads();

    // exclusive prefix (maps strictly before this thread)
    float em = __shfl_up(m, 1);
    float es = __shfl_up(s, 1);
    if (lane == 0) { em = BIGF; es = 0.0f; }         // identity map
    if (wv > 0) {
        float pm = wm[wv - 1], ps = wsum[wv - 1];
        em = fminf(em, pm + es);
        es = ps + es;
    }
    float D0 = fminf(m0, em + s0);
    float D1 = fminf(m1, em + s1);
    float D2 = fminf(m2, em + s2);
    float D3 = fminf(m3, em + s3);
    Dp[j0] = D0; Dp[j0 + 1] = D1; Dp[j0 + 2] = D2; Dp[j0 + 3] = D3;
    __syncthreads();
    return D3;
}

// ---------------------------------------------------------------------------
// Phase 2: DTW over precomputed cost matrix. Single 1024-thread workgroup
// (32 wave32s) on one WGP; D_prev lives in LDS; 3 barriers per row.
// ---------------------------------------------------------------------------
__global__ __launch_bounds__(1024)
void dtw_scan_kernel(const float* __restrict__ c, float* __restrict__ out) {
    __shared__ float Dp[NCOL];
    __shared__ float wm[32];
    __shared__ float wsum[32];
    const int t = threadIdx.x;
    const int lane = t & 31, wv = t >> 5;
    const int j0 = t * 4;
    float result = 0.0f;

    for (int i = 0; i < NROW; ++i) {
        const float* crow = c + (size_t)i * NCOL;
        float4 cv = *(const float4*)(crow + j0);
        if (i + 1 < NROW)                            // global_prefetch_b8 next row
            __builtin_prefetch(crow + NCOL + j0, 0, 0);
        float D3 = dtw_row_scan(cv.x, cv.y, cv.z, cv.w,
                                Dp, wm, wsum, t, lane, wv, j0, i == 0);
        if (i == NROW - 1) result = D3;
    }
    if (j0 + 3 == NCOL - 1) out[0] = result;
}

// ---------------------------------------------------------------------------
// Fallback (only when d_ws cannot hold the 64 MB cost matrix): fuse cost-row
// computation into the scan workgroup. x row cached in LDS; distances
// computed directly as sqrt(sum((x-y)^2)).
// ---------------------------------------------------------------------------
__global__ __launch_bounds__(1024)
void dtw_fused_kernel(const float* __restrict__ x, const float* __restrict__ y,
                      float* __restrict__ out) {
    __shared__ float Dp[NCOL];
    __shared__ float wm[32];
    __shared__ float wsum[32];
    __shared__ float xr[DIMK];
    const int t = threadIdx.x;
    const int lane = t & 31, wv = t >> 5;
    const int j0 = t * 4;
    float result = 0.0f;

    for (int i = 0; i < NROW; ++i) {
        if (t < DIMK) xr[t] = x[(size_t)i * DIMK + t];
        __syncthreads();
        float cd[4];
#pragma unroll
        for (int jj = 0; jj < 4; ++jj) {
            const float* yrow = y + (size_t)(j0 + jj) * DIMK;
            float sq = 0.0f;
#pragma unroll 8
            for (int k = 0; k < DIMK; ++k) {
                float d = xr[k] - yrow[k];
                sq = fmaf(d, d, sq);
            }
            cd[jj] = sqrtf(fmaxf(sq, 1e-12f));
        }
        float D3 = dtw_row_scan(cd[0], cd[1], cd[2], cd[3],
                                Dp, wm, wsum, t, lane, wv, j0, i == 0);
        if (i == NROW - 1) result = D3;
    }
    if (j0 + 3 == NCOL - 1) out[0] = result;
}

// ---------------------------------------------------------------------------
extern "C" void kernel_launch(void* const* d_in, const int* in_sizes, int n_in,
                              void* d_out, int out_size, void* d_ws, size_t ws_size,
                              hipStream_t stream) {
    const float* x = (const float*)d_in[0];
    const float* y = (const float*)d_in[1];
    float* out = (float*)d_out;

    const size_t cbytes = (size_t)NROW * NCOL * sizeof(float);
    const size_t needed = cbytes + (size_t)(NROW + NCOL) * sizeof(float);

    if (ws_size >= needed) {
        float* c  = (float*)d_ws;
        float* xx = c + (size_t)NROW * NCOL;
        float* yy = xx + NROW;
        dtw_norms_kernel<<<(NROW + NCOL + 255) / 256, 256, 0, stream>>>(x, y, xx, yy);
        dtw_cost_wmma_kernel<<<(256 * 256) / 8, 256, 0, stream>>>(x, y, xx, yy, c);
        dtw_scan_kernel<<<1, 1024, 0, stream>>>(c, out);
    } else {
        dtw_fused_kernel<<<1, 1024, 0, stream>>>(x, y, out);
    }
}